// RecurrentGCN_78640851189886
// MI455X (gfx1250) — compile-verified
//
#include <hip/hip_runtime.h>

#define N_NODES 50000
#define N_EDGES 1600000
#define SEQ 64
#define FILTERS 32
#define NTILES (N_NODES / 16)   // 3125, exact
#define WMAT (SEQ * FILTERS)    // 2048 floats per weight matrix

typedef float v2f __attribute__((ext_vector_type(2)));
typedef float v8f __attribute__((ext_vector_type(8)));

static __device__ __forceinline__ v8f wmma_f32_k4(v2f a, v2f b, v8f c) {
    // V_WMMA_F32_16X16X4_F32 : D = A(16x4, f32) * B(4x16, f32) + C(16x16, f32)
    return __builtin_amdgcn_wmma_f32_16x16x4_f32(
        /*neg_a=*/false, a, /*neg_b=*/false, b,
        /*c_mod=*/(short)0, c, /*reuse_a=*/false, /*reuse_b=*/false);
}

// ---------------- workspace zeroing ----------------
__global__ void zero_ws_kernel(float* __restrict__ ws, int n) {
    int i = blockIdx.x * blockDim.x + threadIdx.x;
    if (i < n) ws[i] = 0.0f;
}

// ---------------- degree = segment_sum(w, row) ----------------
__global__ void deg_kernel(const int* __restrict__ row, const float* __restrict__ w,
                           float* __restrict__ deg, int E) {
    int e = blockIdx.x * blockDim.x + threadIdx.x;
    if (e < E) atomicAdd(&deg[row[e]], w[e]);
}

// ---------------- dinv = rsqrt(deg) where deg > 0 ----------------
__global__ void dinv_kernel(const float* __restrict__ deg, float* __restrict__ dinv, int n) {
    int i = blockIdx.x * blockDim.x + threadIdx.x;
    if (i < n) {
        float d = deg[i];
        dinv[i] = (d > 0.0f) ? rsqrtf(d) : 0.0f;
    }
}

// ---------------- T = L_hat @ x  (gather / scale / scatter-add) ----------------
// 16 lanes per edge (2 edges per wave32). Lane 0/16 loads edge metadata and
// computes norm; {r, c, norm} broadcast to the half-wave via shuffles. Each
// lane then moves 4 channels: one float4 gather + 4 global f32 atomics.
// Grid is exact: E*16 threads, 256/block -> no tail, EXEC stays full.
__global__ void spmm_kernel(const int* __restrict__ row, const int* __restrict__ col,
                            const float* __restrict__ w, const float* __restrict__ dinv,
                            const float* __restrict__ x, float* __restrict__ tx1, int E) {
    int t    = blockIdx.x * blockDim.x + threadIdx.x;
    int e    = t >> 4;
    int c4   = t & 15;
    int lane = threadIdx.x & 31;

    int r = 0, c = 0;
    float nrm = 0.0f;
    if (c4 == 0) {                      // lanes 0 and 16: one edge each
        r   = row[e];
        c   = col[e];
        nrm = -dinv[r] * w[e] * dinv[c];
    }
    int src = lane & 16;                // broadcast source lane of this half-wave
    r   = __shfl(r,   src, 32);
    c   = __shfl(c,   src, 32);
    nrm = __shfl(nrm, src, 32);

    const float4* xv = reinterpret_cast<const float4*>(x + (size_t)r * SEQ);
    float4 v = xv[c4];
    float* dst = tx1 + (size_t)c * SEQ + c4 * 4;
    atomicAdd(dst + 0, nrm * v.x);
    atomicAdd(dst + 1, nrm * v.y);
    atomicAdd(dst + 2, nrm * v.z);
    atomicAdd(dst + 3, nrm * v.w);
}

// ---------------- fused WMMA GEMM + GRU epilogue ----------------
// One wave per 16-row tile; 4 weight matrices staged in LDS per block.
//   acc_z = X@Wz0 + T@Wz1 ; acc_h = X@Wh0 + T@Wh1   (N = 32 -> two 16-col tiles)
//   Z = sigmoid(acc_z + bz), Ht = tanh(acc_h + bh), Hn = (1-Z)*Ht
//   out = relu(Hn) @ lin_w + lin_b
__global__ void __launch_bounds__(256)
gru_wmma_kernel(const float* __restrict__ x, const float* __restrict__ tx1,
                const float* __restrict__ wz0, const float* __restrict__ wz1,
                const float* __restrict__ bxz, const float* __restrict__ bhz,
                const float* __restrict__ wh0, const float* __restrict__ wh1,
                const float* __restrict__ bxh, const float* __restrict__ bhh,
                const float* __restrict__ lin_w, const float* __restrict__ lin_b,
                float* __restrict__ out, int ntiles) {
    __shared__ float s_wz0[WMAT];
    __shared__ float s_wz1[WMAT];
    __shared__ float s_wh0[WMAT];
    __shared__ float s_wh1[WMAT];

    // Stage weights (32 KB) into LDS: every wave participates before any exit.
    for (int idx = threadIdx.x; idx < WMAT; idx += 256) {
        s_wz0[idx] = wz0[idx];
        s_wz1[idx] = wz1[idx];
        s_wh0[idx] = wh0[idx];
        s_wh1[idx] = wh1[idx];
    }
    __syncthreads();

    int wave = (blockIdx.x * blockDim.x + threadIdx.x) >> 5;
    int lane = threadIdx.x & 31;
    if (wave >= ntiles) return;   // uniform per wave -> EXEC all-ones for WMMA

    int mlo = lane & 15;          // A: M row / B,C,D: N column (within tile)
    int hi  = lane >> 4;          // A/B: K-half selector ; C/D: M-half selector

    const float* xrow = x   + (size_t)(wave * 16 + mlo) * SEQ + 2 * hi;
    const float* trow = tx1 + (size_t)(wave * 16 + mlo) * SEQ + 2 * hi;

    v8f accz0 = {0.f,0.f,0.f,0.f,0.f,0.f,0.f,0.f};
    v8f accz1 = accz0, acch0 = accz0, acch1 = accz0;

    const int c0 = mlo;        // columns of N-tile 0
    const int c1 = 16 + mlo;   // columns of N-tile 1

    #pragma unroll
    for (int k0 = 0; k0 < SEQ; k0 += 4) {
        // A fragments (16x4 f32): lane holds (M=mlo, K=k0+2*hi+{0,1})
        v2f ax = *reinterpret_cast<const v2f*>(xrow + k0);
        v2f at = *reinterpret_cast<const v2f*>(trow + k0);

        int kr = (k0 + 2 * hi) * FILTERS;   // row-major [64][32] weights
        // B fragments (4x16 f32) from LDS: lane holds (K=k0+2*hi+{0,1}, N=col)
        v2f bz0a = { s_wz0[kr + c0], s_wz0[kr + FILTERS + c0] };
        v2f bz0b = { s_wz0[kr + c1], s_wz0[kr + FILTERS + c1] };
        v2f bz1a = { s_wz1[kr + c0], s_wz1[kr + FILTERS + c0] };
        v2f bz1b = { s_wz1[kr + c1], s_wz1[kr + FILTERS + c1] };
        v2f bh0a = { s_wh0[kr + c0], s_wh0[kr + FILTERS + c0] };
        v2f bh0b = { s_wh0[kr + c1], s_wh0[kr + FILTERS + c1] };
        v2f bh1a = { s_wh1[kr + c0], s_wh1[kr + FILTERS + c0] };
        v2f bh1b = { s_wh1[kr + c1], s_wh1[kr + FILTERS + c1] };

        accz0 = wmma_f32_k4(ax, bz0a, accz0);
        accz0 = wmma_f32_k4(at, bz1a, accz0);
        accz1 = wmma_f32_k4(ax, bz0b, accz1);
        accz1 = wmma_f32_k4(at, bz1b, accz1);
        acch0 = wmma_f32_k4(ax, bh0a, acch0);
        acch0 = wmma_f32_k4(at, bh1a, acch0);
        acch1 = wmma_f32_k4(ax, bh0b, acch1);
        acch1 = wmma_f32_k4(at, bh1b, acch1);
    }

    // Per-column constants (column = c0 for tile 0, c1 for tile 1)
    float bz_0 = bxz[c0] + bhz[c0];
    float bz_1 = bxz[c1] + bhz[c1];
    float bh_0 = bxh[c0] + bhh[c0];
    float bh_1 = bxh[c1] + bhh[c1];
    float lw0  = lin_w[c0];
    float lw1  = lin_w[c1];
    float lb   = lin_b[0];

    // C/D layout: VGPR v holds row M = v + 8*hi, column N = mlo (per N-tile).
    #pragma unroll
    for (int v = 0; v < 8; ++v) {
        float z0 = 1.0f / (1.0f + __expf(-(accz0[v] + bz_0)));
        float z1 = 1.0f / (1.0f + __expf(-(accz1[v] + bz_1)));
        float h0 = tanhf(acch0[v] + bh_0);
        float h1 = tanhf(acch1[v] + bh_1);
        float p  = fmaxf((1.0f - z0) * h0, 0.0f) * lw0
                 + fmaxf((1.0f - z1) * h1, 0.0f) * lw1;
        // reduce the 16 lanes of this half-wave (columns 0..15 of both tiles)
        p += __shfl_xor(p, 1, 32);
        p += __shfl_xor(p, 2, 32);
        p += __shfl_xor(p, 4, 32);
        p += __shfl_xor(p, 8, 32);
        if (mlo == 0) out[wave * 16 + v + 8 * hi] = p + lb;
    }
}

extern "C" void kernel_launch(void* const* d_in, const int* in_sizes, int n_in,
                              void* d_out, int out_size, void* d_ws, size_t ws_size,
                              hipStream_t stream) {
    const float* x    = (const float*)d_in[0];
    const int*   ei   = (const int*)d_in[1];       // edge_index [2, E]
    const float* ew   = (const float*)d_in[2];
    const float* wz0  = (const float*)d_in[3];     // wx_z0
    const float* wz1  = (const float*)d_in[4];     // wx_z1
    const float* bxz  = (const float*)d_in[5];     // bx_z
    const float* bhz  = (const float*)d_in[8];     // bh_z
    // r-gate (indices 9..14) is dead: H=0 makes H*R=0
    const float* wh0  = (const float*)d_in[15];    // wx_h0
    const float* wh1  = (const float*)d_in[16];    // wx_h1
    const float* bxh  = (const float*)d_in[17];    // bx_h
    const float* bhh  = (const float*)d_in[20];    // bh_h
    const float* lw   = (const float*)d_in[21];    // lin_w
    const float* lb   = (const float*)d_in[22];    // lin_b
    float* out = (float*)d_out;

    const int E = N_EDGES;
    const int* row = ei;
    const int* col = ei + E;

    // workspace: [deg: N][dinv: N][tx1: N*64]
    float* deg  = (float*)d_ws;
    float* dinv = deg + N_NODES;
    float* tx1  = dinv + N_NODES;

    int zn = N_NODES * (2 + SEQ);
    zero_ws_kernel<<<(zn + 255) / 256, 256, 0, stream>>>((float*)d_ws, zn);

    deg_kernel<<<(E + 255) / 256, 256, 0, stream>>>(row, ew, deg, E);

    dinv_kernel<<<(N_NODES + 255) / 256, 256, 0, stream>>>(deg, dinv, N_NODES);

    long long spmm_threads = (long long)E * 16;   // 25.6M, divisible by 256
    spmm_kernel<<<(unsigned)(spmm_threads / 256), 256, 0, stream>>>(
        row, col, ew, dinv, x, tx1, E);

    int waves_per_block = 256 / 32;
    int blocks = (NTILES + waves_per_block - 1) / waves_per_block;
    gru_wmma_kernel<<<blocks, 256, 0, stream>>>(
        x, tx1, wz0, wz1, bxz, bhz, wh0, wh1, bxh, bhh, lw, lb, out, NTILES);
}